// ISNEFeaturePropagation_67379446940400
// MI455X (gfx1250) — compile-verified
//
#include <hip/hip_runtime.h>
#include <hip/hip_bf16.h>

#define N_NODES 100000
#define N_EDGES 1250000
#define IN_F 128
#define OUT_F 64

typedef __attribute__((ext_vector_type(2))) float v2f;
typedef __attribute__((ext_vector_type(4))) float v4f;
typedef __attribute__((ext_vector_type(8))) float v8f;

// ---------------------------------------------------------------------------
// Kernel 0: zero the scatter accumulator (d_out) and the per-node edge counts.
// ---------------------------------------------------------------------------
__global__ __launch_bounds__(256) void isne_zero_kernel(float* __restrict__ accum,
                                                        float* __restrict__ count) {
    int tid = blockIdx.x * blockDim.x + threadIdx.x;
    if (tid < N_NODES * OUT_F) accum[tid] = 0.0f;
    if (tid < N_NODES)         count[tid] = 0.0f;
}

// ---------------------------------------------------------------------------
// Kernel 1: h = x @ W via V_WMMA_F32_16X16X4_F32 (exact f32 math, matches ref).
// Block = 256 threads = 8 wave32 waves. Each wave computes a 16x64 strip
// (4 accumulator tiles of 16x16). W (128x64 f32 = 32KB) staged in LDS.
//
// A-frag (16x4 f32, per ISA layout): lanes 0-15 hold K=k,k+1 for row M=lane;
// lanes 16-31 hold K=k+2,k+3 -> one aligned float2 load per lane per k-step.
// B-frag (4x16 f32): VGPR0 = rows K=k (lanes 0-15) / K=k+2 (lanes 16-31),
// VGPR1 = K=k+1 / K=k+3, N = lane&15 -> four LDS scalar reads per tile pair.
// C/D tile: VGPR i -> M = i + 8*(lane>=16), N = lane&15.
// ---------------------------------------------------------------------------
__global__ __launch_bounds__(256) void isne_gemm_kernel(const float* __restrict__ x,
                                                        const float* __restrict__ w,
                                                        float* __restrict__ h) {
    __shared__ float sW[IN_F * OUT_F]; // 32 KB of the 320 KB WGP LDS

    { // cooperative stage of W into LDS (float4, fully coalesced)
        const v4f* src = (const v4f*)w;
        v4f* dst = (v4f*)sW;
        const int n4 = IN_F * OUT_F / 4; // 2048
        for (int i = threadIdx.x; i < n4; i += 256) dst[i] = src[i];
    }
    __syncthreads();

    const int wave = threadIdx.x >> 5;
    const int lane = threadIdx.x & 31;
    const int half = lane >> 4;   // 0: K=k,k+1 ; 1: K=k+2,k+3
    const int ml   = lane & 15;   // M (for A) / N (for B,C,D)

    const int row_base = blockIdx.x * 128 + wave * 16;

    // clamp A-row so OOB waves read valid memory (stores are guarded below)
    int arow = row_base + ml;
    if (arow >= N_NODES) arow = N_NODES - 1;
    const float* xrow = x + (long)arow * IN_F + 2 * half;

    v8f acc0 = {}, acc1 = {}, acc2 = {}, acc3 = {};

    #pragma unroll 4
    for (int k = 0; k < IN_F; k += 4) {
        v2f a = *(const v2f*)(xrow + k); // aligned 8B load: K=k+2h, k+2h+1

        const float* wk = sW + (k + 2 * half) * OUT_F + ml;
        v2f b0, b1, b2, b3;
        b0.x = wk[0];   b0.y = wk[OUT_F + 0];
        b1.x = wk[16];  b1.y = wk[OUT_F + 16];
        b2.x = wk[32];  b2.y = wk[OUT_F + 32];
        b3.x = wk[48];  b3.y = wk[OUT_F + 48];

        acc0 = __builtin_amdgcn_wmma_f32_16x16x4_f32(false, a, false, b0, (short)0, acc0, false, false);
        acc1 = __builtin_amdgcn_wmma_f32_16x16x4_f32(false, a, false, b1, (short)0, acc1, false, false);
        acc2 = __builtin_amdgcn_wmma_f32_16x16x4_f32(false, a, false, b2, (short)0, acc2, false, false);
        acc3 = __builtin_amdgcn_wmma_f32_16x16x4_f32(false, a, false, b3, (short)0, acc3, false, false);
    }

    // store D tiles: VGPR i -> row = row_base + i + 8*half, col = tile*16 + ml
    #pragma unroll
    for (int i = 0; i < 8; i++) {
        int row = row_base + i + 8 * half;
        if (row < N_NODES) {
            float* hr = h + (long)row * OUT_F + ml;
            hr[0]  = acc0[i];
            hr[16] = acc1[i];
            hr[32] = acc2[i];
            hr[48] = acc3[i];
        }
    }
}

// ---------------------------------------------------------------------------
// Kernel 2: edge scatter. One wave32 per edge; lane L handles features 2L,2L+1.
// Gather of h[src] is a coalesced 256B float2 read per wave; accumulation via
// global f32 atomics (L2-resident: accum fits in the 192MB L2). Edge scalars
// are wave-uniform -> scalarized loads.
// ---------------------------------------------------------------------------
__global__ __launch_bounds__(256) void isne_scatter_kernel(const float* __restrict__ h,
                                                           const float* __restrict__ ew,
                                                           const long long* __restrict__ ei,
                                                           float* __restrict__ accum,
                                                           float* __restrict__ count) {
    long long tid = (long long)blockIdx.x * 256 + threadIdx.x;
    long long e = tid >> 5;
    int lane = (int)(tid & 31);
    if (e >= N_EDGES) return;

    int src = (int)ei[e];            // edge_index row 0
    int dst = (int)ei[N_EDGES + e];  // edge_index row 1
    float wgt = ew[e];

    v2f hv = *(const v2f*)(h + (long long)src * OUT_F + lane * 2);
    float* ap = accum + (long long)dst * OUT_F + lane * 2;
    atomicAdd(ap + 0, hv.x * wgt);
    atomicAdd(ap + 1, hv.y * wgt);
    if (lane == 0) atomicAdd(count + dst, 1.0f);
}

// ---------------------------------------------------------------------------
// Kernel 3: out = relu(accum / max(count,1) + bias), in place over d_out.
// ---------------------------------------------------------------------------
__global__ __launch_bounds__(256) void isne_finalize_kernel(float* __restrict__ out,
                                                            const float* __restrict__ count,
                                                            const float* __restrict__ bias) {
    int tid = blockIdx.x * 256 + threadIdx.x;
    if (tid >= N_NODES * OUT_F) return;
    int n = tid >> 6;
    int f = tid & 63;
    float c = fmaxf(count[n], 1.0f);
    float v = out[tid] / c + bias[f];
    out[tid] = fmaxf(v, 0.0f);
}

extern "C" void kernel_launch(void* const* d_in, const int* in_sizes, int n_in,
                              void* d_out, int out_size, void* d_ws, size_t ws_size,
                              hipStream_t stream) {
    (void)in_sizes; (void)n_in; (void)out_size; (void)ws_size;

    const float*     x           = (const float*)d_in[0];      // [N_NODES, IN_F]
    const float*     weight      = (const float*)d_in[1];      // [IN_F, OUT_F]
    const float*     bias        = (const float*)d_in[2];      // [OUT_F]
    const float*     edge_weight = (const float*)d_in[3];      // [N_EDGES]
    const long long* edge_index  = (const long long*)d_in[4];  // [2, N_EDGES] int64

    float* out = (float*)d_out;                                 // doubles as accumulator
    float* h   = (float*)d_ws;                                  // [N_NODES, OUT_F] = 25.6 MB
    float* cnt = (float*)((char*)d_ws + (size_t)N_NODES * OUT_F * sizeof(float)); // [N_NODES]

    // 0) zero accumulator + counts
    {
        int total  = N_NODES * OUT_F;
        int blocks = (total + 255) / 256;
        isne_zero_kernel<<<blocks, 256, 0, stream>>>(out, cnt);
    }
    // 1) h = x @ W  (WMMA f32)
    {
        int blocks = (N_NODES + 127) / 128; // 128 rows per block (8 waves x 16 rows)
        isne_gemm_kernel<<<blocks, 256, 0, stream>>>(x, weight, h);
    }
    // 2) edge gather/scale/scatter-add
    {
        long long threads = (long long)N_EDGES * 32;
        int blocks = (int)((threads + 255) / 256);
        isne_scatter_kernel<<<blocks, 256, 0, stream>>>(h, edge_weight, edge_index, out, cnt);
    }
    // 3) mean + bias + relu
    {
        int total  = N_NODES * OUT_F;
        int blocks = (total + 255) / 256;
        isne_finalize_kernel<<<blocks, 256, 0, stream>>>(out, cnt, bias);
    }
}